// CBP_44392781972147
// MI455X (gfx1250) — compile-verified
//
#include <hip/hip_runtime.h>

// Compact Bilinear Pooling, direct sparse-convolution formulation.
// B=8, C=512, W=512, D=8192. One workgroup per output row (b,w); the
// D-length circular-convolution accumulator lives in LDS (32 KB).

#define CBP_B 8
#define CBP_C 512
#define CBP_W 512
#define CBP_D 8192
#define CBP_THREADS 256

typedef float __attribute__((ext_vector_type(4))) v4f;

#if defined(__gfx1250__) && __has_builtin(__builtin_amdgcn_global_load_async_to_lds_b32)
#define CBP_ASYNC 1
#else
#define CBP_ASYNC 0
#endif

#if CBP_ASYNC
typedef __attribute__((address_space(1))) int* cbp_gptr;
typedef __attribute__((address_space(3))) int* cbp_lptr;
__device__ __forceinline__ void cbp_async_b32(const void* g, void* l) {
  // (global src, LDS dst, imm offset, imm cpol) -> GLOBAL_LOAD_ASYNC_TO_LDS_B32
  __builtin_amdgcn_global_load_async_to_lds_b32((cbp_gptr)g, (cbp_lptr)l, 0, 0);
}
#endif

__device__ __forceinline__ void cbp_wait_async() {
#if defined(__gfx1250__)
#if __has_builtin(__builtin_amdgcn_s_wait_asynccnt)
  __builtin_amdgcn_s_wait_asynccnt(0);
#else
  asm volatile("s_wait_asynccnt 0" ::: "memory");
#endif
#endif
}

__global__ __launch_bounds__(CBP_THREADS) void cbp_kernel(
    const float* __restrict__ x1, const float* __restrict__ x2,
    const int* __restrict__ h1, const float* __restrict__ s1,
    const int* __restrict__ h2, const float* __restrict__ s2,
    float* __restrict__ out) {
  __shared__ __align__(16) float acc[CBP_D];   // 32 KB circular-conv accumulator
  __shared__ float2 tab[CBP_C];                // packed (y2[j], h2[j] as float bits)
#if CBP_ASYNC
  __shared__ float xraw[CBP_C];
  __shared__ float s2s[CBP_C];
  __shared__ int   h2s[CBP_C];
#endif

  const int r = blockIdx.x;            // row index in [0, B*W): r = b*W + w
  const int b = r >> 9;                // W == 512
  const int w = r & (CBP_W - 1);
  const int t = threadIdx.x;

  const float* x1row = x1 + (size_t)b * (CBP_C * CBP_W) + w;  // channel stride = W
  const float* x2row = x2 + (size_t)b * (CBP_C * CBP_W) + w;

#if CBP_ASYNC
  // Stage operand-2 row + sketch tables into LDS with async DMA copies.
  for (int c = t; c < CBP_C; c += CBP_THREADS) {
    cbp_async_b32(x2row + (size_t)c * CBP_W, &xraw[c]);
    cbp_async_b32(s2 + c, &s2s[c]);
    cbp_async_b32(h2 + c, &h2s[c]);
  }
#endif

  // Zero the accumulator while the async copies are in flight.
  for (int d = t; d < CBP_D; d += CBP_THREADS) acc[d] = 0.0f;

  // Each thread owns two operand-1 channels in registers: signed value + bin.
  const int i0 = t;
  const int i1 = t + CBP_THREADS;
  float y1a = x1row[(size_t)i0 * CBP_W] * s1[i0];
  float y1b = x1row[(size_t)i1 * CBP_W] * s1[i1];
  int   ha  = h1[i0];
  int   hb  = h1[i1];

#if CBP_ASYNC
  cbp_wait_async();
  __syncthreads();
  for (int c = t; c < CBP_C; c += CBP_THREADS)
    tab[c] = make_float2(xraw[c] * s2s[c], __int_as_float(h2s[c]));
#else
  for (int c = t; c < CBP_C; c += CBP_THREADS)
    tab[c] = make_float2(x2row[(size_t)c * CBP_W] * s2[c], __int_as_float(h2[c]));
#endif

  __syncthreads();

  // Scattered outer product: out[(h1[i]+h2[j]) & (D-1)] += y1[i]*y2[j].
  // tab[j] is a wave-uniform broadcast read (single ds_load_b64, no conflicts).
#pragma unroll 8
  for (int j = 0; j < CBP_C; ++j) {
    float2 p = tab[j];
    float v = p.x;
    int   o = __float_as_int(p.y);
    atomicAdd(&acc[(ha + o) & (CBP_D - 1)], y1a * v);  // ds_add_f32 (no return)
    atomicAdd(&acc[(hb + o) & (CBP_D - 1)], y1b * v);
  }

  __syncthreads();

  // Stream the 32 KB row out with non-temporal 128-bit stores.
  v4f* __restrict__ o4 = (v4f*)(out + (size_t)r * CBP_D);
  const v4f* a4 = (const v4f*)acc;
  for (int q = t; q < CBP_D / 4; q += CBP_THREADS)
    __builtin_nontemporal_store(a4[q], o4 + q);
}

extern "C" void kernel_launch(void* const* d_in, const int* in_sizes, int n_in,
                              void* d_out, int out_size, void* d_ws, size_t ws_size,
                              hipStream_t stream) {
  (void)in_sizes; (void)n_in; (void)d_ws; (void)ws_size; (void)out_size;
  const float* x1 = (const float*)d_in[0];   // bottom1 [B, C, W] f32
  const float* x2 = (const float*)d_in[1];   // bottom2 [B, C, W] f32
  const int*   h1 = (const int*)d_in[2];     // rand_h_1 [C]
  const float* s1 = (const float*)d_in[3];   // rand_s_1 [C]
  const int*   h2 = (const int*)d_in[4];     // rand_h_2 [C]
  const float* s2 = (const float*)d_in[5];   // rand_s_2 [C]
  float* out = (float*)d_out;                // [B*W, D] f32 viewed as [B, 512, D]

  dim3 grid(CBP_B * CBP_W);                  // 4096 rows, one block each
  cbp_kernel<<<grid, CBP_THREADS, 0, stream>>>(x1, x2, h1, s1, h2, s2, out);
}